// GCNGraphClassifier_88648124990850
// MI455X (gfx1250) — compile-verified
//
#include <hip/hip_runtime.h>
#include <hip/hip_bf16.h>

typedef __attribute__((ext_vector_type(2))) float v2f;
typedef __attribute__((ext_vector_type(8))) float v8f;

#define HIDDIM 128
#define NCLASSES 10
#define NGRAPHS 512

// ---------------------------------------------------------------------------
// Degree: deg[dst[e]] += 1  (self loop added later as +1 in rsqrt)
// ---------------------------------------------------------------------------
__global__ void deg_count_kernel(const int* __restrict__ dst, float* __restrict__ deg,
                                 int nedges) {
    int e = blockIdx.x * blockDim.x + threadIdx.x;
    if (e < nedges) atomicAdd(&deg[dst[e]], 1.0f);
}

__global__ void make_dinv_kernel(const float* __restrict__ deg, float* __restrict__ dinv,
                                 int nnodes) {
    int i = blockIdx.x * blockDim.x + threadIdx.x;
    if (i < nnodes) dinv[i] = rsqrtf(deg[i] + 1.0f);   // deg includes self-loop -> always > 0
}

// ---------------------------------------------------------------------------
// WMMA GEMM: out[nrows,128] = A[gather(rows),128] @ W[128,128]
// idx == nullptr -> identity row gather.
// One block = 256 threads = 8 waves. Block computes a 16x128 output slab.
// Wave w computes the 16x16 tile at columns [16w, 16w+16).
// K=128 -> 32 x V_WMMA_F32_16X16X4_F32 per tile.
// ---------------------------------------------------------------------------
#define SA_STRIDE 132   // 128 + 4 floats pad: lane->bank stride of 4, conflict-free

__global__ __launch_bounds__(256) void gemm128_wmma_kernel(
        const float* __restrict__ A, const int* __restrict__ idx,
        const float* __restrict__ W, float* __restrict__ out, int nrows) {
    __shared__ float sA[16 * SA_STRIDE];   // 8.25 KB

    const int tid = threadIdx.x;
    const int m0  = blockIdx.x * 16;

    // Stage A slab (with optional embedding gather): 16 rows x 128 cols = 512 float4
    for (int i = tid; i < 16 * 32; i += 256) {
        const int r  = i >> 5;
        const int c4 = i & 31;
        const int row = m0 + r;
        float4 v = make_float4(0.f, 0.f, 0.f, 0.f);
        if (row < nrows) {
            const long srow = idx ? (long)idx[row] : (long)row;
            v = ((const float4*)(A + srow * (long)HIDDIM))[c4];
        }
        ((float4*)(sA + r * SA_STRIDE))[c4] = v;
    }
    __syncthreads();

    const int wave = tid >> 5;          // 0..7 -> column tile
    const int lane = tid & 31;
    const int half = lane >> 4;         // 0: lanes 0-15, 1: lanes 16-31
    const int lm   = lane & 15;
    const int n0   = wave * 16;

    v8f c = {};
    #pragma unroll
    for (int kk = 0; kk < 32; ++kk) {
        const int k0 = kk * 4;
        // A 16x4 f32 layout: lanes 0-15 -> M=lane, {v0,v1}={K0,K1}; lanes 16-31 -> {K2,K3}
        v2f a, b;
        a.x = sA[lm * SA_STRIDE + k0 + 2 * half + 0];
        a.y = sA[lm * SA_STRIDE + k0 + 2 * half + 1];
        // B 4x16 f32 layout: v0 holds K rows {0,2}, v1 holds {1,3}; N = lm
        b.x = W[(k0 + 2 * half + 0) * HIDDIM + n0 + lm];
        b.y = W[(k0 + 2 * half + 1) * HIDDIM + n0 + lm];
        c = __builtin_amdgcn_wmma_f32_16x16x4_f32(
                /*neg_a=*/false, a, /*neg_b=*/false, b,
                /*c_mod=*/(short)0, c, /*reuse_a=*/false, /*reuse_b=*/false);
    }

    // C/D 16x16 f32: VGPR i -> row m0 + i + 8*half, col n0 + lm
    #pragma unroll
    for (int i = 0; i < 8; ++i) {
        const int row = m0 + half * 8 + i;
        if (row < nrows) out[(long)row * HIDDIM + n0 + lm] = c[i];
    }
}

// ---------------------------------------------------------------------------
// Edge scatter: agg[dst] += h[src] * (dinv[src]*dinv[dst]); one wave per edge
// ---------------------------------------------------------------------------
__global__ __launch_bounds__(256) void edge_scatter_kernel(
        const float* __restrict__ h, const int* __restrict__ src,
        const int* __restrict__ dst, const float* __restrict__ dinv,
        float* __restrict__ agg, int nedges) {
    const int e = blockIdx.x * 8 + (threadIdx.x >> 5);
    const int lane = threadIdx.x & 31;
    if (e >= nedges) return;
    const int s = src[e];
    const int d = dst[e];
    const float norm = dinv[s] * dinv[d];
    const float4 v = ((const float4*)(h + (long)s * HIDDIM))[lane];
    float* ad = agg + (long)d * HIDDIM + lane * 4;
    atomicAdd(&ad[0], v.x * norm);
    atomicAdd(&ad[1], v.y * norm);
    atomicAdd(&ad[2], v.z * norm);
    atomicAdd(&ad[3], v.w * norm);
}

// ---------------------------------------------------------------------------
// Finalize: x = relu(agg + h*dinv^2 + bias)   (self-loop term fused here)
// ---------------------------------------------------------------------------
__global__ void finalize_kernel(const float* __restrict__ agg, const float* __restrict__ h,
                                const float* __restrict__ dinv, const float* __restrict__ bias,
                                float* __restrict__ out, int total) {
    const int i = blockIdx.x * blockDim.x + threadIdx.x;
    if (i >= total) return;
    const int node = i >> 7;
    const int col  = i & (HIDDIM - 1);
    const float di = dinv[node];
    const float v  = agg[i] + h[i] * di * di + bias[col];
    out[i] = v > 0.f ? v : 0.f;
}

// ---------------------------------------------------------------------------
// Mean-pool accumulation
// ---------------------------------------------------------------------------
__global__ void pool_accum_kernel(const float* __restrict__ x, const int* __restrict__ batch,
                                  float* __restrict__ sums, float* __restrict__ cnts,
                                  int total) {
    const int i = blockIdx.x * blockDim.x + threadIdx.x;
    if (i >= total) return;
    const int node = i >> 7;
    const int col  = i & (HIDDIM - 1);
    const int g = batch[node];
    atomicAdd(&sums[g * HIDDIM + col], x[i]);
    if (col == 0) atomicAdd(&cnts[g], 1.0f);
}

// ---------------------------------------------------------------------------
// Final linear: out[g,c] = (sum_k sums[g,k]*linW[k,c]) / max(cnt,1) + linb[c]
// ---------------------------------------------------------------------------
__global__ void final_linear_kernel(const float* __restrict__ sums, const float* __restrict__ cnts,
                                    const float* __restrict__ linW, const float* __restrict__ linb,
                                    float* __restrict__ out) {
    const int g = blockIdx.x;
    const int c = threadIdx.x;
    if (c >= NCLASSES) return;
    float cnt = cnts[g];
    cnt = cnt > 1.f ? cnt : 1.f;
    float acc = 0.f;
    #pragma unroll 8
    for (int k = 0; k < HIDDIM; ++k) acc += sums[g * HIDDIM + k] * linW[k * NCLASSES + c];
    out[g * NCLASSES + c] = acc / cnt + linb[c];
}

// ---------------------------------------------------------------------------
extern "C" void kernel_launch(void* const* d_in, const int* in_sizes, int n_in,
                              void* d_out, int out_size, void* d_ws, size_t ws_size,
                              hipStream_t stream) {
    const int*   x_idx = (const int*)d_in[0];
    const int*   eidx  = (const int*)d_in[1];
    const int*   batch = (const int*)d_in[2];
    const float* emb   = (const float*)d_in[3];
    const float* W1    = (const float*)d_in[4];
    const float* b1    = (const float*)d_in[5];
    const float* W2    = (const float*)d_in[6];
    const float* b2    = (const float*)d_in[7];
    const float* linW  = (const float*)d_in[8];
    const float* linb  = (const float*)d_in[9];

    const int N = in_sizes[0];           // 100000 nodes
    const int E = in_sizes[1] / 2;       // 1600000 edges
    const int* src = eidx;
    const int* dst = eidx + E;

    const long feat = (long)N * HIDDIM;  // 12.8M floats

    float* bufH = (float*)d_ws;          // [N,128] linear output
    float* bufA = bufH + feat;           // [N,128] aggregation accumulator
    float* bufX = bufA + feat;           // [N,128] layer activations
    float* deg  = bufX + feat;           // [N]
    float* dinv = deg + N;               // [N]
    float* sums = dinv + N;              // [512,128]
    float* cnts = sums + NGRAPHS * HIDDIM; // [512]

    // ---- degree / normalization ----
    hipMemsetAsync(deg, 0, (size_t)N * sizeof(float), stream);
    deg_count_kernel<<<(E + 255) / 256, 256, 0, stream>>>(dst, deg, E);
    make_dinv_kernel<<<(N + 255) / 256, 256, 0, stream>>>(deg, dinv, N);

    const int gemm_blocks = (N + 15) / 16;
    const int elem_blocks = (int)((feat + 255) / 256);

    // ---- layer 1: h = emb[x_idx] @ W1 ; aggregate ; relu(+b1) ----
    gemm128_wmma_kernel<<<gemm_blocks, 256, 0, stream>>>(emb, x_idx, W1, bufH, N);
    hipMemsetAsync(bufA, 0, (size_t)feat * sizeof(float), stream);
    edge_scatter_kernel<<<(E + 7) / 8, 256, 0, stream>>>(bufH, src, dst, dinv, bufA, E);
    finalize_kernel<<<elem_blocks, 256, 0, stream>>>(bufA, bufH, dinv, b1, bufX, (int)feat);

    // ---- layer 2: h = x @ W2 ; aggregate ; relu(+b2) ----
    gemm128_wmma_kernel<<<gemm_blocks, 256, 0, stream>>>(bufX, nullptr, W2, bufH, N);
    hipMemsetAsync(bufA, 0, (size_t)feat * sizeof(float), stream);
    edge_scatter_kernel<<<(E + 7) / 8, 256, 0, stream>>>(bufH, src, dst, dinv, bufA, E);
    finalize_kernel<<<elem_blocks, 256, 0, stream>>>(bufA, bufH, dinv, b2, bufX, (int)feat);

    // ---- global mean pool + final linear ----
    hipMemsetAsync(sums, 0, (size_t)(NGRAPHS * HIDDIM + NGRAPHS) * sizeof(float), stream);
    pool_accum_kernel<<<elem_blocks, 256, 0, stream>>>(bufX, batch, sums, cnts, (int)feat);
    final_linear_kernel<<<NGRAPHS, 32, 0, stream>>>(sums, cnts, linW, linb, (float*)d_out);
}